// knn_up_6201932775995
// MI455X (gfx1250) — compile-verified
//
#include <hip/hip_runtime.h>

typedef __attribute__((ext_vector_type(2))) float v2f;
typedef __attribute__((ext_vector_type(8))) float v8f;

// ---------------------------------------------------------------------------
// Kernel 0: pack voxel coords into float4 (x, y, z, -0.5*||v||^2)
// ---------------------------------------------------------------------------
__global__ void knn_pack_vox(const float* __restrict__ vc,
                             float4* __restrict__ packed, int M) {
    int i = blockIdx.x * blockDim.x + threadIdx.x;
    if (i < M) {
        float x = vc[3 * i + 0];
        float y = vc[3 * i + 1];
        float z = vc[3 * i + 2];
        packed[i] = make_float4(x, y, z, -0.5f * (x * x + y * y + z * z));
    }
}

// Branchless sorted-triple insert (s0 >= s1 >= s2, larger score = nearer).
__device__ __forceinline__ void insert3(float v, int m,
                                        float& s0, float& s1, float& s2,
                                        int& i0, int& i1, int& i2) {
    bool g0 = v > s0;
    bool g1 = v > s1;
    bool g2 = v > s2;
    float ns0 = g0 ? v : s0;
    int ni0 = g0 ? m : i0;
    float ns1 = g0 ? s0 : (g1 ? v : s1);
    int ni1 = g0 ? i0 : (g1 ? m : i1);
    float ns2 = g1 ? s1 : (g2 ? v : s2);
    int ni2 = g1 ? i1 : (g2 ? m : i2);
    s0 = ns0; s1 = ns1; s2 = ns2;
    i0 = ni0; i1 = ni1; i2 = ni2;
}

// ---------------------------------------------------------------------------
// Kernel 1: WMMA distance scan + per-lane top-3.
// One wave32 handles 16 points; loops over all voxels 16 at a time.
//   A (16x4, voxels):  lane<16: VGPR0=x, VGPR1=y ; lane>=16: VGPR0=z, VGPR1=-0.5||v||^2
//   B (4x16, points):  VGPR0: lanes0-15 = px (K=0), lanes16-31 = pz (K=2)
//                      VGPR1: lanes0-15 = py (K=1), lanes16-31 = 1.0 (K=3)
//   D (16x16): VGPR r, lane l: score(voxel = base + r + 8*(l>=16), point = l%16)
// => score = v.p - 0.5||v||^2 ; maximizing it == minimizing d^2.
//
// The top-3 insert chain only runs when the ballot says some lane's new
// 8-value max beats its current 3rd-best: ~P(any lane) ~ min(1, 96/t), so
// only a few hundred of the 2048 iterations pay the full VALU cost.
// ---------------------------------------------------------------------------
__global__ void knn_scan(const float4* __restrict__ vpk,
                         const float* __restrict__ pc,
                         int* __restrict__ out_idx,
                         float* __restrict__ out_w,
                         int M, int N) {
    const int lane = threadIdx.x & 31;
    const int wave = threadIdx.x >> 5;
    const int pbase = (blockIdx.x * (blockDim.x >> 5) + wave) << 4;
    const bool hi = lane >= 16;
    const int l16 = lane & 15;

    // B operand (constant over the voxel loop)
    const int pidx = pbase + l16;
    const float px = pc[3 * pidx + 0];
    const float py = pc[3 * pidx + 1];
    const float pz = pc[3 * pidx + 2];
    v2f b;
    b[0] = hi ? pz : px;
    b[1] = hi ? 1.0f : py;

    float s0 = -3.4e38f, s1 = -3.4e38f, s2 = -3.4e38f;
    int i0 = 0, i1 = 0, i2 = 0;

    for (int vb = 0; vb < M; vb += 16) {
        // prefetch a few tiles ahead on the streaming voxel array
        __builtin_prefetch(&vpk[vb + 128], 0, 0);

        float4 q = vpk[vb + l16];
        v2f a;
        a[0] = hi ? q.z : q.x;
        a[1] = hi ? q.w : q.y;

        v8f c = {};
        c = __builtin_amdgcn_wmma_f32_16x16x4_f32(
                /*neg_a=*/false, a, /*neg_b=*/false, b,
                /*c_mod=*/(short)0, c, /*reuse_a=*/false, /*reuse_b=*/false);

        // Cheap guard: max of the 8 new scores vs current 3rd best.
        float m8 = fmaxf(fmaxf(fmaxf(c[0], c[1]), fmaxf(c[2], c[3])),
                         fmaxf(fmaxf(c[4], c[5]), fmaxf(c[6], c[7])));

        // Wave-uniform branch (EXEC stays all-ones around the WMMA).
        if (__ballot(m8 > s2)) {
            const int mb = vb + (hi ? 8 : 0);
#pragma unroll
            for (int r = 0; r < 8; ++r) {
                insert3(c[r], mb + r, s0, s1, s2, i0, i1, i2);
            }
        }
    }

    // Merge the two halves of the lane pair (lane l <-> lane l+16):
    // both halves hold top-3 candidates for the same point l%16.
    {
        float os0 = __shfl_xor(s0, 16, 32);
        float os1 = __shfl_xor(s1, 16, 32);
        float os2 = __shfl_xor(s2, 16, 32);
        int oi0 = __shfl_xor(i0, 16, 32);
        int oi1 = __shfl_xor(i1, 16, 32);
        int oi2 = __shfl_xor(i2, 16, 32);
        insert3(os0, oi0, s0, s1, s2, i0, i1, i2);
        insert3(os1, oi1, s0, s1, s2, i0, i1, i2);
        insert3(os2, oi2, s0, s1, s2, i0, i1, i2);
    }

    if (!hi) {
        // Recompute exact squared distances on the 3 winners (matches reference),
        // then inverse-distance weights, normalized.
        int idx[3] = {i0, i1, i2};
        float w[3];
        float wsum = 0.0f;
#pragma unroll
        for (int j = 0; j < 3; ++j) {
            float4 q = vpk[idx[j]];
            float dx = q.x - px;
            float dy = q.y - py;
            float dz = q.z - pz;
            float d2 = dx * dx + dy * dy + dz * dz;
            float wi = 1.0f / (d2 + 1e-8f);
            w[j] = wi;
            wsum += wi;
        }
        float inv = 1.0f / wsum;
#pragma unroll
        for (int j = 0; j < 3; ++j) {
            out_idx[3 * pidx + j] = idx[j];
            out_w[3 * pidx + j] = w[j] * inv;
        }
    }
}

// ---------------------------------------------------------------------------
// Kernel 2: weighted feature gather. One thread per 4 output features.
// ---------------------------------------------------------------------------
__global__ void knn_gather(const float* __restrict__ vf,
                           const int* __restrict__ idx,
                           const float* __restrict__ w,
                           float* __restrict__ out,
                           int N, int D4total, int D) {
    int g = blockIdx.x * blockDim.x + threadIdx.x;
    if (g >= D4total) return;
    const int d4 = D >> 2;
    int n = g / d4;
    int c = (g - n * d4) << 2;

    int j0 = idx[3 * n + 0], j1 = idx[3 * n + 1], j2 = idx[3 * n + 2];
    float w0 = w[3 * n + 0], w1 = w[3 * n + 1], w2 = w[3 * n + 2];

    const float4 f0 = *(const float4*)(vf + (size_t)j0 * D + c);
    const float4 f1 = *(const float4*)(vf + (size_t)j1 * D + c);
    const float4 f2 = *(const float4*)(vf + (size_t)j2 * D + c);

    float4 r;
    r.x = w0 * f0.x + w1 * f1.x + w2 * f2.x;
    r.y = w0 * f0.y + w1 * f1.y + w2 * f2.y;
    r.z = w0 * f0.z + w1 * f1.z + w2 * f2.z;
    r.w = w0 * f0.w + w1 * f1.w + w2 * f2.w;
    *(float4*)(out + (size_t)n * D + c) = r;
}

extern "C" void kernel_launch(void* const* d_in, const int* in_sizes, int n_in,
                              void* d_out, int out_size, void* d_ws, size_t ws_size,
                              hipStream_t stream) {
    const float* v_coor = (const float*)d_in[0];   // [M,3]
    const float* v_feats = (const float*)d_in[1];  // [M,D]
    const float* p_coor = (const float*)d_in[2];   // [N,3]
    float* out = (float*)d_out;

    const int M = in_sizes[0] / 3;
    const int D = in_sizes[1] / M;
    const int N = in_sizes[2] / 3;

    // Workspace layout: packed voxels | idx | weights
    float4* packed = (float4*)d_ws;
    int* widx = (int*)((char*)d_ws + (size_t)M * sizeof(float4));
    float* wgt = (float*)((char*)d_ws + (size_t)M * sizeof(float4) +
                          (size_t)N * 3 * sizeof(int));

    knn_pack_vox<<<(M + 255) / 256, 256, 0, stream>>>(v_coor, packed, M);

    // 256 threads = 8 waves; each wave handles 16 points -> 128 points/block.
    knn_scan<<<N / 128, 256, 0, stream>>>(packed, p_coor, widx, wgt, M, N);

    const int D4total = N * (D / 4);
    knn_gather<<<(D4total + 255) / 256, 256, 0, stream>>>(v_feats, widx, wgt, out,
                                                          N, D4total, D);
}